// DenseSIFTDescriptor_8272107012369
// MI455X (gfx1250) — compile-verified
//
#include <hip/hip_runtime.h>
#include <hip/hip_bf16.h>
#include <math.h>
#include <stdint.h>

// DenseSIFT fused kernel for MI455X (gfx1250, wave32).
//
// Roofline: 256MB f32 output write dominates (23.3 TB/s -> ~11us floor);
// all intermediate stages live in LDS. The identity "reshape" conv is a
// gather; the separable 4-tap triangular pooling is done with
// V_WMMA_F32_16X16X4_F32 (banded weight matrix x data tile, 5 K-chunks
// per 16-wide band). One wave owns one angular channel so the
// horizontal->vertical pooling dependency is wave-local (LDS in-order).
// Banded weights are built once in registers (cndmask selects, no EXEC
// manipulation) and shared by all 40 WMMAs.

#define NBINS 8
#define TILE  16
#define H     512
#define W     512

typedef __attribute__((ext_vector_type(2))) float v2f;
typedef __attribute__((ext_vector_type(8))) float v8f;

#if defined(__has_builtin)
#  if __has_builtin(__builtin_amdgcn_wmma_f32_16x16x4_f32)
#    define HAVE_WMMA_F32X4 1
#  endif
#  if __has_builtin(__builtin_amdgcn_global_load_async_to_lds_b32)
#    define HAVE_ASYNC_LDS 1
#  endif
#  if __has_builtin(__builtin_amdgcn_s_wait_asynccnt)
#    define HAVE_WAIT_ASYNC 1
#  endif
#endif

__device__ __forceinline__ int imin(int a, int b) { return a < b ? a : b; }
__device__ __forceinline__ int imax(int a, int b) { return a > b ? a : b; }

__device__ __forceinline__ void async_copy_f32(const float* g, float* l) {
#ifdef HAVE_ASYNC_LDS
  // Builtin expects addrspace(1)/addrspace(3) int*. generic->AS via integer
  // cast: low 32 bits of a generic LDS address are the allocation-relative
  // LDS offset (ISA 10.2 aperture rules); global generic == AS1 numerically.
  __builtin_amdgcn_global_load_async_to_lds_b32(
      (__attribute__((address_space(1))) int*)(uintptr_t)g,
      (__attribute__((address_space(3))) int*)(uint32_t)(uintptr_t)l,
      0, 0);
#else
  *l = *g;
#endif
}

__device__ __forceinline__ void wait_async_copies() {
#ifdef HAVE_ASYNC_LDS
#  ifdef HAVE_WAIT_ASYNC
  __builtin_amdgcn_s_wait_asynccnt(0);
#  else
  asm volatile("s_wait_asynccnt 0" ::: "memory");
#  endif
#endif
}

// Banded-weight element: w[d] for d in [0,3], else 0. Pure VALU select chain.
__device__ __forceinline__ float band_w(int d, float w0, float w1, float w2,
                                        float w3) {
  float r = (d == 0) ? w0 : 0.0f;
  r = (d == 1) ? w1 : r;
  r = (d == 2) ? w2 : r;
  r = (d == 3) ? w3 : r;
  return r;
}

__global__ __launch_bounds__(256) void dsift_kernel(
    const float* __restrict__ x,   // (2,1,512,512)
    const float* __restrict__ pk,  // (1,1,4,4) separable triangular kernel
    float* __restrict__ out)       // (2,128,512,512)
{
  // LDS tiles (output tile = 16x16 pixels):
  //  xs : x rows i0-4..i0+19 (clamped)           24 x 24
  //  ang: angular hist, ang rows i0-3..i0+18     8 x 22 x 22 (zero outside img)
  //  hp : horizontally pooled, cols j0-1..j0+17  8 x 22 x 19
  //  vp : fully pooled, rows/cols  -1..+17 rel.  8 x 19 x 19
  __shared__ float xs[24][25];
  __shared__ float ang[NBINS][22][23];
  __shared__ float hp[NBINS][22][20];
  __shared__ float vp[NBINS][19][20];
  __shared__ float wsep[4];

  const int t  = threadIdx.x;
  const int b  = blockIdx.z;
  const int i0 = blockIdx.y * TILE;
  const int j0 = blockIdx.x * TILE;
  const size_t xbase = (size_t)b * (size_t)(H * W);

  // Separable weights: pool = v v^T with v > 0 -> v[k] = sqrt(diag[k]).
  if (t < 4) wsep[t] = sqrtf(fmaxf(pk[t * 4 + t], 0.0f));

  // ---- Stage 0: stage x tile into LDS (async global->LDS path) ----
  for (int e = t; e < 24 * 24; e += 256) {
    int r = e / 24, c = e % 24;
    int gr = imin(imax(i0 - 4 + r, 0), H - 1);   // replicate clamp == edge pad
    int gc = imin(imax(j0 - 4 + c, 0), W - 1);
    async_copy_f32(x + xbase + (size_t)gr * W + gc, &xs[r][c]);
  }
  wait_async_copies();
  __syncthreads();

  // ---- Stage 1: gradients -> soft angular binning (zero-pad outside image) --
  const float TWO_PI = 6.28318530717958647692f;
  const float BIN_SCALE = (float)NBINS / TWO_PI;
  for (int e = t; e < 22 * 22; e += 256) {
    int r = e / 22, c = e % 22;
    int ir = i0 - 3 + r, jc = j0 - 3 + c;
#pragma unroll
    for (int k = 0; k < NBINS; ++k) ang[k][r][c] = 0.0f;
    if ((unsigned)ir < (unsigned)H && (unsigned)jc < (unsigned)W) {
      int rl = r + 1, cl = c + 1;  // xs-local coords of (ir,jc)
      int cp = imin(jc + 1, W - 1) - (j0 - 4);
      int cm = imax(jc - 1, 0)     - (j0 - 4);
      int rp = imin(ir + 1, H - 1) - (i0 - 4);
      int rm = imax(ir - 1, 0)     - (i0 - 4);
      float gx = 0.5f * (xs[rl][cp] - xs[rl][cm]);
      float gy = 0.5f * (xs[rp][cl] - xs[rm][cl]);
      float mag = sqrtf(gx * gx + gy * gy + 1e-10f);
      float ori = atan2f(gy, gx + 1e-10f) + TWO_PI;   // in (pi, 3pi]
      float ob  = ori * BIN_SCALE;                    // in (4, 12]
      float f   = floorf(ob);
      float w1  = ob - f;
      int b0 = ((int)f) & 7;
      int b1 = (b0 + 1) & 7;
      ang[b0][r][c] = (1.0f - w1) * mag;
      ang[b1][r][c] = w1 * mag;
    }
  }
  __syncthreads();

  // ---- Stage 2: separable 4-tap pooling via WMMA f32 16x16x4 ----
  const int wave = t >> 5;
  const int lane = t & 31;
  const int half = lane >> 4;  // K-pair selector in 16x4 / 4x16 f32 layout
  const int lm   = lane & 15;
  const int ch   = wave;       // 8 waves <-> 8 angular channels

#ifdef HAVE_WMMA_F32X4
  // Per-chunk banded weight operands, built once in registers and shared by
  // all 8 WMMA tiles (weights are identical across bands/groups/directions:
  // element (k0+j, lm) of the banded matrix is wsep[k0+j-lm] or 0).
  const float w0 = wsep[0], w1s = wsep[1], w2 = wsep[2], w3 = wsep[3];
  v2f wk[5];
#pragma unroll
  for (int kc = 0; kc < 5; ++kc) {
    int k0 = 4 * kc + 2 * half;
    wk[kc].x = band_w(k0 - lm,     w0, w1s, w2, w3);
    wk[kc].y = band_w(k0 + 1 - lm, w0, w1s, w2, w3);
  }

  // Horizontal: hp[y][n] = sum_r ang[y][g+r] * Wb[r][n], Wb[r][n]=wx[r-n].
  // A = ang data (16x4 chunk), B = banded weights. 2 row-bands x 2 col-groups
  // x 5 K-chunks cover the 22 x 19 hp tile.
#pragma unroll
  for (int band = 0; band <= 6; band += 6) {
#pragma unroll
    for (int g = 0; g <= 3; g += 3) {
      v8f acc = {0.f, 0.f, 0.f, 0.f, 0.f, 0.f, 0.f, 0.f};
#pragma unroll
      for (int kc = 0; kc < 5; ++kc) {
        int k0 = 4 * kc + 2 * half;          // K index of VGPR0 for this lane
        v2f a;
        a.x = ang[ch][band + lm][imin(g + k0,     21)];
        a.y = ang[ch][band + lm][imin(g + k0 + 1, 21)];
        acc = __builtin_amdgcn_wmma_f32_16x16x4_f32(
            false, a, false, wk[kc], (short)0, acc, false, false);
      }
#pragma unroll
      for (int v = 0; v < 8; ++v)
        hp[ch][band + v + 8 * half][g + lm] = acc[v];
    }
  }
  // Vertical: vp[m][n] = sum_r Wb[m][r] * hp[br+r][n], Wb[m][r]=wy[r-m].
  // hp was written by this same wave (same channel); LDS is in-order per wave.
#pragma unroll
  for (int br = 0; br <= 3; br += 3) {
#pragma unroll
    for (int g = 0; g <= 3; g += 3) {
      v8f acc = {0.f, 0.f, 0.f, 0.f, 0.f, 0.f, 0.f, 0.f};
#pragma unroll
      for (int kc = 0; kc < 5; ++kc) {
        int k0 = 4 * kc + 2 * half;
        v2f bm;
        bm.x = hp[ch][imin(br + k0,     21)][g + lm];
        bm.y = hp[ch][imin(br + k0 + 1, 21)][g + lm];
        acc = __builtin_amdgcn_wmma_f32_16x16x4_f32(
            false, wk[kc], false, bm, (short)0, acc, false, false);
      }
#pragma unroll
      for (int v = 0; v < 8; ++v)
        vp[ch][br + v + 8 * half][g + lm] = acc[v];
    }
  }
#else
  // VALU fallback (kept so a missing builtin still compiles).
  for (int e = t; e < NBINS * 22 * 19; e += 256) {
    int d = e / (22 * 19), rem = e % (22 * 19);
    int y = rem / 19, xx = rem % 19;
    float s = 0.f;
#pragma unroll
    for (int k = 0; k < 4; ++k) s += wsep[k] * ang[d][y][xx + k];
    hp[d][y][xx] = s;
  }
  __syncthreads();
  for (int e = t; e < NBINS * 19 * 19; e += 256) {
    int d = e / (19 * 19), rem = e % (19 * 19);
    int yy = rem / 19, xx = rem % 19;
    float s = 0.f;
#pragma unroll
    for (int k = 0; k < 4; ++k) s += wsep[k] * hp[d][yy + k][xx];
    vp[d][yy][xx] = s;
  }
#endif
  __syncthreads();

  // ---- Stage 3: gather 128-vec, L2 -> clip(0.2) -> L2 -> RootSIFT, store ----
  const int il = t >> 4, jl = t & 15;
  const int i = i0 + il, j = j0 + jl;
  bool rv[4], cv[4];
#pragma unroll
  for (int k = 0; k < 4; ++k) {
    rv[k] = (unsigned)(i + k - 1) <= (unsigned)H;  // pooled row in [0,512]
    cv[k] = (unsigned)(j + k - 1) <= (unsigned)W;  // pooled col in [0,512]
  }

  float s1 = 0.0f;
  for (int d = 0; d < NBINS; ++d)
#pragma unroll
    for (int ky = 0; ky < 4; ++ky)
#pragma unroll
      for (int kx = 0; kx < 4; ++kx) {
        float v = (rv[ky] && cv[kx]) ? vp[d][il + ky][jl + kx] : 0.0f;
        s1 += v * v;
      }
  float inv1 = 1.0f / fmaxf(sqrtf(s1), 1e-12f);

  float sc2 = 0.0f, sab = 0.0f;
  for (int d = 0; d < NBINS; ++d)
#pragma unroll
    for (int ky = 0; ky < 4; ++ky)
#pragma unroll
      for (int kx = 0; kx < 4; ++kx) {
        float v = (rv[ky] && cv[kx]) ? vp[d][il + ky][jl + kx] : 0.0f;
        float cc = fminf(fmaxf(v * inv1, 0.0f), 0.2f);
        sc2 += cc * cc;
        sab += cc;  // values are non-negative -> |.| == .
      }
  float n2 = fmaxf(sqrtf(sc2), 1e-12f);
  float l1 = fmaxf(sab / n2, 1e-12f);
  float invd = 1.0f / (n2 * l1);

  const size_t obase =
      (size_t)b * 128u * (size_t)(H * W) + (size_t)i * W + j;
  for (int d = 0; d < NBINS; ++d)
#pragma unroll
    for (int ky = 0; ky < 4; ++ky)
#pragma unroll
      for (int kx = 0; kx < 4; ++kx) {
        float v = (rv[ky] && cv[kx]) ? vp[d][il + ky][jl + kx] : 0.0f;
        float cc = fminf(fmaxf(v * inv1, 0.0f), 0.2f);
        int c = d * 16 + ky * 4 + kx;
        // 256MB stream > 192MB L2: bypass with non-temporal stores.
        __builtin_nontemporal_store(sqrtf(cc * invd + 1e-10f),
                                    out + obase + (size_t)c * (H * W));
      }
}

extern "C" void kernel_launch(void* const* d_in, const int* in_sizes, int n_in,
                              void* d_out, int out_size, void* d_ws, size_t ws_size,
                              hipStream_t stream) {
  (void)in_sizes; (void)n_in; (void)out_size; (void)d_ws; (void)ws_size;
  const float* x  = (const float*)d_in[0];
  const float* pk = (const float*)d_in[1];
  // d_in[2] (reshape_kernel) is eye(128).reshape(128,8,4,4) by construction:
  // the conv it drives is a pure im2col gather, implemented directly above.
  float* out = (float*)d_out;
  dim3 grid(W / TILE, H / TILE, 2);
  dim3 block(256);
  dsift_kernel<<<grid, block, 0, stream>>>(x, pk, out);
}